// LMNLayer_730144441068
// MI455X (gfx1250) — compile-verified
//
#include <hip/hip_runtime.h>
#include <cmath>

// LMN layer, MI455X (gfx1250, wave32).
// IN = HID = MEM = 1024, T = 512, B = 64, all fp32.
//
// Per step t (two stream-ordered kernels, hard dependency chain):
//   h      = tanh( x[t] @ Wxh^T + m @ Wmh^T + bh )   (kernel A, 2 GEMMs)
//   m_new  =        h  @ Whm^T + m @ Wmm^T + bm      (kernel B, 2 GEMMs -> d_out[t])
// m for step t is d_out[t-1] (or m_prev at t=0); m_final copied at the end.
//
// GEMM core: V_WMMA_F32_16X16X4_F32, M=64, N=1024, K=1024 per GEMM.
// Block = 128 threads (4 waves) computes a 64(M) x 32(N) tile; each wave owns
// a 16-row strip with two 16x16 f32 accumulators. Weight slabs (32 N-rows x
// 32 K) are DOUBLE-BUFFERED in LDS and filled with the CDNA5 async path
// (global_load_async_to_lds_b128, ASYNCcnt) so the fill for chunk k+1 is
// hidden behind the WMMA stream for chunk k. Padded LDS stride (36 floats)
// keeps the ds_load_b64 fragment reads bank-conflict free.

#define NT 512
#define NB 64
#define DIM 1024

typedef __attribute__((ext_vector_type(2))) float v2f;
typedef __attribute__((ext_vector_type(8))) float v8f;

#define KC  32   // K chunk staged in LDS per iteration
#define BN  32   // N tile per block
#define PAD 36   // padded LDS row stride in floats (mult of 4 -> 16B aligned rows)

// Issue this thread's share (2 x b128) of an async global->LDS fill of the
// W slab [n0..n0+BN-1] x [k0..k0+KC-1] into `buf`. Tracked by ASYNCcnt.
__device__ __forceinline__ void fill_slab_async(const float* __restrict__ W,
                                                float* buf, int n0, int k0,
                                                int tid)
{
#pragma unroll
  for (int i = 0; i < (BN * KC) / (128 * 4); ++i) {  // 2 per thread
    int idx = tid + 128 * i;            // 0..255
    int row = idx >> 3;                 // 0..31
    int col = (idx & 7) << 2;           // 0,4,..,28 (floats)
    // Low 32 bits of a generic LDS pointer == byte offset into LDS
    // (flat LDS aperture).
    unsigned loff = (unsigned)(size_t)(buf + row * PAD + col);
    const float* g = W + (size_t)(n0 + row) * DIM + k0 + col;
    asm volatile("global_load_async_to_lds_b128 %0, %1, off"
                 :: "v"(loff), "v"(g)
                 : "memory");
  }
}

// acc[0..1] += A[m0..m0+15, :] x W^T[:, n0..n0+31] via v_wmma_f32_16x16x4_f32.
// W is row-major [N, K] (so B[k][n] = W[n][k]). buf0/buf1: LDS double buffer.
__device__ __forceinline__ void gemm_acc(const float* __restrict__ A,
                                         const float* __restrict__ W,
                                         float* buf0, float* buf1, v8f* acc,
                                         int m0, int n0, int l16, int lhi,
                                         int tid)
{
  // Prologue: start the fill for chunk 0.
  fill_slab_async(W, buf0, n0, 0, tid);

  for (int k0 = 0; k0 < DIM; k0 += KC) {
    const bool odd = (k0 / KC) & 1;
    float* cur = odd ? buf1 : buf0;
    float* nxt = odd ? buf0 : buf1;

    // All waves finished reading `nxt` (two chunks ago) -> safe to overwrite.
    __syncthreads();
    if (k0 + KC < DIM) {
      fill_slab_async(W, nxt, n0, k0 + KC, tid);
      // ASYNCcnt is in-order: <=2 outstanding means `cur`'s fills landed.
      asm volatile("s_wait_asynccnt 0x2" ::: "memory");
    } else {
      asm volatile("s_wait_asynccnt 0x0" ::: "memory");
    }
    // Make every wave's fills of `cur` visible to all waves.
    __syncthreads();

    // A fragment base: row m0+l16, K pair starting at 2*lhi.
    // 16x4 f32 A layout: VGPR0 = K{0 | 2}, VGPR1 = K{1 | 3} (lanes 0-15 | 16-31).
    const float* arow = A + (size_t)(m0 + l16) * DIM + k0 + 2 * lhi;
#pragma unroll
    for (int kk = 0; kk < KC; kk += 4) {
      v2f a = *(const v2f*)(arow + kk);
#pragma unroll
      for (int j = 0; j < 2; ++j) {
        // 4x16 f32 B layout mirrors A: lane holds column n0+16j+l16,
        // VGPR pair holds K{2*lhi, 2*lhi+1}.
        v2f b = *(const v2f*)(cur + (16 * j + l16) * PAD + kk + 2 * lhi);
        acc[j] = __builtin_amdgcn_wmma_f32_16x16x4_f32(
            false, a, false, b, (short)0, acc[j], false, false);
      }
    }
  }
}

template <bool TANH>
__global__ __launch_bounds__(128) void lmn_step_gemm(
    const float* __restrict__ A1, const float* __restrict__ W1,
    const float* __restrict__ A2, const float* __restrict__ W2,
    const float* __restrict__ bias, float* __restrict__ out)
{
  __shared__ float lds[2 * BN * PAD];  // double buffer, 9216 B
  float* buf0 = lds;
  float* buf1 = lds + BN * PAD;

  const int tid = threadIdx.x;
  const int lane = tid & 31;
  const int l16 = lane & 15;
  const int lhi = lane >> 4;           // 0 or 1
  const int m0 = (tid >> 5) << 4;      // wave id * 16
  const int n0 = blockIdx.x * BN;

  v8f acc[2];
#pragma unroll
  for (int j = 0; j < 2; ++j)
#pragma unroll
    for (int r = 0; r < 8; ++r) acc[j][r] = 0.0f;

  gemm_acc(A1, W1, buf0, buf1, acc, m0, n0, l16, lhi, tid);
  gemm_acc(A2, W2, buf0, buf1, acc, m0, n0, l16, lhi, tid);

  // C/D 16x16 f32 layout: VGPR r -> row m0 + r + 8*lhi, lane -> col n0+16j+l16.
#pragma unroll
  for (int j = 0; j < 2; ++j) {
    const int col = n0 + 16 * j + l16;
    const float bv = bias[col];
#pragma unroll
    for (int r = 0; r < 8; ++r) {
      const int row = m0 + r + 8 * lhi;
      float v = acc[j][r] + bv;
      if (TANH) v = tanhf(v);
      out[(size_t)row * DIM + col] = v;
    }
  }
}

__global__ void copy_f32(const float* __restrict__ src, float* __restrict__ dst,
                         int n)
{
  int i = blockIdx.x * blockDim.x + threadIdx.x;
  if (i < n) dst[i] = src[i];
}

extern "C" void kernel_launch(void* const* d_in, const int* in_sizes, int n_in,
                              void* d_out, int out_size, void* d_ws,
                              size_t ws_size, hipStream_t stream)
{
  (void)in_sizes; (void)n_in; (void)out_size; (void)ws_size;

  const float* x      = (const float*)d_in[0];  // [T, B, IN]
  const float* m_prev = (const float*)d_in[1];  // [B, MEM]
  const float* Wxh    = (const float*)d_in[2];  // [HID, IN]
  const float* Whm    = (const float*)d_in[3];  // [MEM, HID]
  const float* Wmm    = (const float*)d_in[4];  // [MEM, MEM]
  const float* Wmh    = (const float*)d_in[5];  // [HID, MEM]
  const float* bh     = (const float*)d_in[6];  // [HID]
  const float* bm     = (const float*)d_in[7];  // [MEM]

  float* outs = (float*)d_out;  // [T, B, MEM] then m_final [B, MEM]
  float* h    = (float*)d_ws;   // [B, HID] scratch (256 KB)

  dim3 grid(DIM / BN);   // 32 blocks
  dim3 block(128);       // 4 wave32s

  for (int t = 0; t < NT; ++t) {
    const float* m = (t == 0) ? m_prev : outs + (size_t)(t - 1) * NB * DIM;
    lmn_step_gemm<true><<<grid, block, 0, stream>>>(
        x + (size_t)t * NB * DIM, Wxh, m, Wmh, bh, h);
    lmn_step_gemm<false><<<grid, block, 0, stream>>>(
        h, Whm, m, Wmm, bm, outs + (size_t)t * NB * DIM);
  }

  // m_final = outs[T-1]
  const int n = NB * DIM;
  copy_f32<<<n / 256, 256, 0, stream>>>(
      outs + (size_t)(NT - 1) * NB * DIM, outs + (size_t)NT * NB * DIM, n);
}